// CollisionChecker_70377334112311
// MI455X (gfx1250) — compile-verified
//
#include <hip/hip_runtime.h>

typedef __attribute__((ext_vector_type(2))) float v2f;
typedef __attribute__((ext_vector_type(8))) float v8f;

#define B_N 128
#define T_N 256
#define H_N 256
#define W_N 256
#define C_N 16

// One block per batch b, 256 threads = 8 wave32s.
// Each wave handles 32 points as two 16-point WMMA tiles:
//   mass(point) = sum over 16 channels, computed as D = ones(16x4) x B(4x16) + C
//   accumulated over 4 channel-groups via V_WMMA_F32_16X16X4_F32.
__global__ __launch_bounds__(256) void CollisionChecker_70377334112311_kernel(
    const float* __restrict__ traj,   // (B, T, 2)
    const float* __restrict__ amap,   // (B, H, W, C)
    float* __restrict__ valid)        // (B,)
{
    __shared__ int s_wave_ok[8];

    const int b    = blockIdx.x;
    const int tid  = threadIdx.x;
    const int lane = tid & 31;
    const int wave = tid >> 5;

    const float* batch_traj = traj + (size_t)b * (T_N * 2);
    const float* batch_map  = amap + (size_t)b * ((size_t)H_N * W_N * C_N);

    // A matrix = all ones (16x4 f32 -> 2 VGPRs, every element 1.0)
    v2f a; a.x = 1.0f; a.y = 1.0f;

    // B layout (4x16 f32, 2 VGPRs): lanes 0-15 carry K={g*4+0, g*4+1} of
    // column N=lane; lanes 16-31 carry K={g*4+2, g*4+3} of column N=lane-16.
    const int chanSel = (lane >= 16) ? 2 : 0;

    int wave_ok = 1;

    #pragma unroll
    for (int tile = 0; tile < 2; ++tile) {
        const int point = wave * 32 + tile * 16 + (lane & 15);

        const float x = batch_traj[point * 2 + 0];
        const float y = batch_traj[point * 2 + 1];

        // Match reference: grid = int32(((pt + 10)/20) * 256), trunc toward 0
        const float fx = ((x + 10.0f) / 20.0f) * 256.0f;
        const float fy = ((y + 10.0f) / 20.0f) * 256.0f;
        const int gi = (int)fx;
        const int gj = (int)fy;

        const bool inb = (gi >= 0) && (gi < H_N) && (gj >= 0) && (gj < W_N);
        const int gic = min(max(gi, 0), H_N - 1);
        const int gjc = min(max(gj, 0), W_N - 1);

        // Base of this point's 16-channel (64B-aligned) vector
        const float* p = batch_map + ((size_t)gic * W_N + gjc) * C_N + chanSel;

        // 4 x global_load_b64: channel pairs for channel groups g=0..3
        v2f b0 = *(const v2f*)(p + 0);
        v2f b1 = *(const v2f*)(p + 4);
        v2f b2 = *(const v2f*)(p + 8);
        v2f b3 = *(const v2f*)(p + 12);

        // D[m][n] = sum_k B[k][n] accumulated over 4 groups -> mass(point n)
        v8f c = {};
        c = __builtin_amdgcn_wmma_f32_16x16x4_f32(false, a, false, b0,
                                                  (short)0, c, false, false);
        c = __builtin_amdgcn_wmma_f32_16x16x4_f32(false, a, false, b1,
                                                  (short)0, c, false, false);
        c = __builtin_amdgcn_wmma_f32_16x16x4_f32(false, a, false, b2,
                                                  (short)0, c, false, false);
        c = __builtin_amdgcn_wmma_f32_16x16x4_f32(false, a, false, b3,
                                                  (short)0, c, false, false);

        // C/D layout: VGPR0 of lane l holds D[., N=l&15] -> mass of our point
        const float mass = c[0];

        const bool ok = inb && !(mass > 100.0f);

        // wave32 ballot: all 32 lanes active; lanes l and l+16 duplicate the
        // same point's predicate, which is AND-neutral.
        const unsigned long long bal = __ballot(ok);
        wave_ok &= (bal == 0xFFFFFFFFull) ? 1 : 0;
    }

    if (lane == 0) s_wave_ok[wave] = wave_ok;
    __syncthreads();

    if (tid == 0) {
        int all_ok = 1;
        #pragma unroll
        for (int w = 0; w < 8; ++w) all_ok &= s_wave_ok[w];
        valid[b] = all_ok ? 1.0f : 0.0f;
    }
}

extern "C" void kernel_launch(void* const* d_in, const int* in_sizes, int n_in,
                              void* d_out, int out_size, void* d_ws, size_t ws_size,
                              hipStream_t stream) {
    const float* traj = (const float*)d_in[0];  // (128, 256, 2) f32
    const float* amap = (const float*)d_in[1];  // (128, 256, 256, 16) f32
    float* valid = (float*)d_out;               // (128,) bool -> float 0/1

    dim3 grid(B_N);
    dim3 block(256);
    CollisionChecker_70377334112311_kernel<<<grid, block, 0, stream>>>(traj, amap, valid);
}